// MultiHeadAttentionBlock_41059887349881
// MI455X (gfx1250) — compile-verified
//
#include <hip/hip_runtime.h>
#include <math.h>

typedef __attribute__((ext_vector_type(2))) float v2f;
typedef __attribute__((ext_vector_type(8))) float v8f;
typedef int v4i __attribute__((vector_size(16)));   // b128 async payload type

#define B_SZ      16
#define NQH       32
#define NKVH      8
#define HD        128
#define DIM       4096
#define T_LEN     4096
#define KVDIM     (NKVH * HD)       // 1024
#define WAVES_PB  8

// gfx1250 async global->LDS path (ASYNCcnt), guarded so the build never breaks.
#if defined(__has_builtin)
#  if __has_builtin(__builtin_amdgcn_global_load_async_to_lds_b128) && \
      __has_builtin(__builtin_amdgcn_s_wait_asynccnt)
#    define USE_ASYNC_LDS 1
#  endif
#endif

// ---------------------------------------------------------------------------
// C[16 x N] = A[16 x K] * W[N x K]^T   (i.e. x @ W.T), f32 WMMA 16x16x4.
// One block (8 waves) per 16-column tile; each wave owns a K/8 slice with
// scalar (readfirstlane) loop control so EXEC stays all-ones around WMMA;
// unroll x8 batches 16 b64 loads ahead of 8 back-to-back v_wmma ops.
// ---------------------------------------------------------------------------
template <int K>
__global__ void wmma_gemm16_nt(const float* __restrict__ A,
                               const float* __restrict__ W,
                               float* __restrict__ C,
                               int N) {
    __shared__ float red[WAVES_PB][32][8];
    const int tid  = threadIdx.x;
    const int wave = __builtin_amdgcn_readfirstlane(tid >> 5);  // wave-uniform SGPR
    const int lane = tid & 31;                                  // wave32
    const int n0   = blockIdx.x * 16;

    constexpr int kslice = K / WAVES_PB;
    const int kbeg = wave * kslice;

    // A 16x4 f32 layout: lanes 0-15 -> M=lane, K={k,k+1}; lanes 16-31 -> K={k+2,k+3}
    const int row = lane & 15;
    const int klo = (lane >> 4) << 1;

    const float* arow = A + (size_t)row * K + kbeg + klo;
    const float* wrow = W + (size_t)(n0 + row) * K + kbeg + klo;  // B[kk][n] = W[n][kk]

    v8f acc = {};
#pragma unroll 8
    for (int k = 0; k < kslice; k += 4) {
        v2f a, b;
        a.x = arow[k];
        a.y = arow[k + 1];
        b.x = wrow[k];
        b.y = wrow[k + 1];
        // (neg_a, A, neg_b, B, c_mod, C, reuse_a, reuse_b)
        acc = __builtin_amdgcn_wmma_f32_16x16x4_f32(
                  false, a, false, b, (short)0, acc, false, false);
    }

#pragma unroll
    for (int j = 0; j < 8; ++j) red[wave][lane][j] = acc[j];
    __syncthreads();

    // 256 threads reduce the 32x8 element tile across the 8 waves.
    const int l = tid >> 3;   // lane slot 0..31
    const int j = tid & 7;    // vgpr slot 0..7
    float s = 0.f;
#pragma unroll
    for (int w = 0; w < WAVES_PB; ++w) s += red[w][l][j];

    const int m = j + ((l >= 16) ? 8 : 0);   // C/D layout: M = j + 8*(lane>=16)
    const int n = n0 + (l & 15);
    C[(size_t)m * N + n] = s;
}

// ---------------------------------------------------------------------------
// In-place RoPE on q[16][32][128] and k[16][8][128] (S=1: one cos/sin row[64]).
// ---------------------------------------------------------------------------
__global__ void rope_kernel(float* __restrict__ q, float* __restrict__ k,
                            const float* __restrict__ cosv,
                            const float* __restrict__ sinv) {
    const int QP = B_SZ * NQH * (HD / 2);   // 32768 pairs
    const int KP = B_SZ * NKVH * (HD / 2);  //  8192 pairs
    int idx = blockIdx.x * blockDim.x + threadIdx.x;
    if (idx >= QP + KP) return;
    float* base;
    int pair;
    if (idx < QP) { base = q; pair = idx; }
    else          { base = k; pair = idx - QP; }
    const int i = pair & 63;                // frequency index within head
    const float c = cosv[i], s = sinv[i];
    const float a = base[2 * pair];
    const float b = base[2 * pair + 1];
    base[2 * pair]     = a * c - b * s;
    base[2 * pair + 1] = a * s + b * c;
}

// ---------------------------------------------------------------------------
// scores[b][h][t] = q[b][h] . key[b][t][h/4] / sqrt(128)
// block = (b,kvh) x 512-position chunk; K-cache read once, reused by 4 heads.
// Position 4095 substitutes the freshly roped k (virtual cache update).
// ---------------------------------------------------------------------------
__global__ void attn_scores(const float* __restrict__ q,     // [16][32][128]
                            const float* __restrict__ knew,  // [16][8][128]
                            const float* __restrict__ ckeys, // [16][4096][8][128]
                            float* __restrict__ scores) {    // [16][32][4096]
    __shared__ float qs[4][HD];
    const int bk   = blockIdx.x;          // 0..127
    const int b    = bk >> 3;
    const int kvh  = bk & 7;
    const int tcb  = blockIdx.y * 512;    // chunk base
    const int tid  = threadIdx.x;         // 256 threads

    for (int i = tid; i < 4 * HD; i += 256)
        qs[i >> 7][i & 127] = q[((size_t)b * NQH + kvh * 4 + (i >> 7)) * HD + (i & 127)];
    __syncthreads();

    const float scale = 0.08838834764831845f;   // 1/sqrt(128)

    for (int tt = 0; tt < 2; ++tt) {
        const int t = tcb + tt * 256 + tid;
        const float* kp = (t == T_LEN - 1)
            ? (knew + ((size_t)b * NKVH + kvh) * HD)
            : (ckeys + (((size_t)b * T_LEN + t) * NKVH + kvh) * HD);
        float d0 = 0.f, d1 = 0.f, d2 = 0.f, d3 = 0.f;
#pragma unroll 8
        for (int dd = 0; dd < HD; dd += 4) {
            const float4 kq = *(const float4*)(kp + dd);
            d0 = fmaf(kq.x, qs[0][dd], fmaf(kq.y, qs[0][dd+1], fmaf(kq.z, qs[0][dd+2], fmaf(kq.w, qs[0][dd+3], d0))));
            d1 = fmaf(kq.x, qs[1][dd], fmaf(kq.y, qs[1][dd+1], fmaf(kq.z, qs[1][dd+2], fmaf(kq.w, qs[1][dd+3], d1))));
            d2 = fmaf(kq.x, qs[2][dd], fmaf(kq.y, qs[2][dd+1], fmaf(kq.z, qs[2][dd+2], fmaf(kq.w, qs[2][dd+3], d2))));
            d3 = fmaf(kq.x, qs[3][dd], fmaf(kq.y, qs[3][dd+1], fmaf(kq.z, qs[3][dd+2], fmaf(kq.w, qs[3][dd+3], d3))));
        }
        const size_t base = ((size_t)b * NQH + kvh * 4) * T_LEN + t;
        scores[base]              = d0 * scale;
        scores[base + T_LEN]      = d1 * scale;
        scores[base + 2 * T_LEN]  = d2 * scale;
        scores[base + 3 * T_LEN]  = d3 * scale;
    }
}

// ---------------------------------------------------------------------------
// In-place softmax over the 4096-wide score rows; one block per (b,h).
// ---------------------------------------------------------------------------
__global__ void softmax4096(float* __restrict__ scores) {
    __shared__ float red[256];
    float* row = scores + (size_t)blockIdx.x * T_LEN;
    const int tid = threadIdx.x;

    float m = -INFINITY;
    for (int t = tid; t < T_LEN; t += 256) m = fmaxf(m, row[t]);
    red[tid] = m;
    __syncthreads();
    for (int s = 128; s > 0; s >>= 1) {
        if (tid < s) red[tid] = fmaxf(red[tid], red[tid + s]);
        __syncthreads();
    }
    m = red[0];
    __syncthreads();

    float sum = 0.f;
    for (int t = tid; t < T_LEN; t += 256) {
        const float e = __expf(row[t] - m);
        row[t] = e;
        sum += e;
    }
    red[tid] = sum;
    __syncthreads();
    for (int s = 128; s > 0; s >>= 1) {
        if (tid < s) red[tid] += red[tid + s];
        __syncthreads();
    }
    const float inv = 1.f / red[0];
    for (int t = tid; t < T_LEN; t += 256) row[t] *= inv;
}

// ---------------------------------------------------------------------------
// out[b][kvh*4+h4][d] = sum_t probs * v[t][d]; V-cache streamed once.
// Probability tiles (4 heads x 256 positions = 4KB) are staged into LDS via
// gfx1250 async global->LDS b128 loads (ASYNCcnt) when available.
// ---------------------------------------------------------------------------
__global__ void attn_out(const float* __restrict__ probs,  // [16][32][4096]
                         const float* __restrict__ cvals,  // [16][4096][8][128]
                         const float* __restrict__ vnew,   // [16][8][128]
                         float* __restrict__ out) {        // [16][32][128]
    __shared__ float pl[4][256];
    float* plf = &pl[0][0];
    const int bk  = blockIdx.x;
    const int b   = bk >> 3;
    const int kvh = bk & 7;
    const int tid = threadIdx.x;          // 128 threads = d

    float a0 = 0.f, a1 = 0.f, a2 = 0.f, a3 = 0.f;
    for (int t0 = 0; t0 < T_LEN; t0 += 256) {
        __syncthreads();
        // Stage probs[4][256] tile: thread i owns 8 consecutive floats.
        {
            const int flat = tid * 8;               // 0..1023
            const int h4   = flat >> 8;             // head within group
            const int t    = flat & 255;            // position within tile
            const float* gsrc =
                probs + ((size_t)b * NQH + kvh * 4 + h4) * T_LEN + t0 + t;
#if defined(USE_ASYNC_LDS)
            __builtin_amdgcn_global_load_async_to_lds_b128(
                (__attribute__((address_space(1))) v4i*)gsrc,
                (__attribute__((address_space(3))) v4i*)(plf + flat), 0, 0);
            __builtin_amdgcn_global_load_async_to_lds_b128(
                (__attribute__((address_space(1))) v4i*)gsrc,
                (__attribute__((address_space(3))) v4i*)(plf + flat), 16, 0);
            __builtin_amdgcn_s_wait_asynccnt(0);
#else
#pragma unroll
            for (int u = 0; u < 8; ++u) plf[flat + u] = gsrc[u];
#endif
        }
        __syncthreads();
        for (int t = 0; t < 256; ++t) {
            const int tt = t0 + t;
            const float vv = (tt == T_LEN - 1)
                ? vnew[((size_t)b * NKVH + kvh) * HD + tid]
                : cvals[(((size_t)b * T_LEN + tt) * NKVH + kvh) * HD + tid];
            a0 = fmaf(pl[0][t], vv, a0);
            a1 = fmaf(pl[1][t], vv, a1);
            a2 = fmaf(pl[2][t], vv, a2);
            a3 = fmaf(pl[3][t], vv, a3);
        }
    }
    const size_t ob = ((size_t)b * NQH + kvh * 4) * HD + tid;
    out[ob]          = a0;
    out[ob + HD]     = a1;
    out[ob + 2 * HD] = a2;
    out[ob + 3 * HD] = a3;
}

// ---------------------------------------------------------------------------
extern "C" void kernel_launch(void* const* d_in, const int* in_sizes, int n_in,
                              void* d_out, int out_size, void* d_ws, size_t ws_size,
                              hipStream_t stream) {
    (void)in_sizes; (void)n_in; (void)out_size; (void)ws_size;

    const float* x    = (const float*)d_in[0];
    // d_in[1] = start_pos (fixed 4095 in this problem; shapes are compile-time)
    const float* fcos = (const float*)d_in[2];
    const float* fsin = (const float*)d_in[3];
    const float* wq   = (const float*)d_in[4];
    const float* wk   = (const float*)d_in[5];
    const float* wv   = (const float*)d_in[6];
    const float* wo   = (const float*)d_in[7];
    const float* ck   = (const float*)d_in[8];
    const float* cv   = (const float*)d_in[9];
    float* out = (float*)d_out;

    // workspace layout (floats)
    float* q     = (float*)d_ws;               // 16*4096  = 65536
    float* k     = q + B_SZ * DIM;             // 16*1024  = 16384
    float* v     = k + B_SZ * KVDIM;           // 16*1024  = 16384
    float* attn  = v + B_SZ * KVDIM;           // 16*4096  = 65536
    float* probs = attn + B_SZ * DIM;          // 16*32*4096 = 2097152  (~8.6 MB total)

    // QKV projections (WMMA f32)
    wmma_gemm16_nt<DIM><<<DIM / 16,   256, 0, stream>>>(x, wq, q, DIM);
    wmma_gemm16_nt<DIM><<<KVDIM / 16, 256, 0, stream>>>(x, wk, k, KVDIM);
    wmma_gemm16_nt<DIM><<<KVDIM / 16, 256, 0, stream>>>(x, wv, v, KVDIM);

    // RoPE on q and k (in workspace; inputs untouched)
    rope_kernel<<<(B_SZ * (NQH + NKVH) * (HD / 2) + 255) / 256, 256, 0, stream>>>(
        q, k, fcos, fsin);

    // Attention: scores -> softmax -> weighted V
    dim3 sgrid(B_SZ * NKVH, T_LEN / 512);
    attn_scores<<<sgrid, 256, 0, stream>>>(q, k, ck, probs);
    softmax4096<<<B_SZ * NQH, 256, 0, stream>>>(probs);
    attn_out<<<B_SZ * NKVH, 128, 0, stream>>>(probs, cv, v, attn);

    // Output projection (WMMA f32) straight into d_out
    wmma_gemm16_nt<DIM><<<DIM / 16, 256, 0, stream>>>(attn, wo, out, DIM);
}